// ProFAMEncoder_91207925498128
// MI455X (gfx1250) — compile-verified
//
#include <hip/hip_runtime.h>
#include <hip/hip_bf16.h>

// Problem constants (match reference)
#define B_    8
#define S_    2048
#define D_    1280
#define OUT_  1280
#define K2_   (2 * D_)   // 2560 fused features

// Tiling
#define TM    64
#define TN    64
#define TK    64
#define LDP   68         // LDS row pitch in dwords: 16B-aligned float4 stores,
                         // lane bank stride 4 -> conflict-free ds_load_b64 frag reads

typedef __attribute__((ext_vector_type(2))) float v2f;
typedef __attribute__((ext_vector_type(8))) float v8f;

// ---------------------------------------------------------------------------
// Pass 1: n[b] = number of valid tokens per batch (mask is a prefix mask).
// ---------------------------------------------------------------------------
__global__ __launch_bounds__(256)
void profam_count_kernel(const unsigned char* __restrict__ mask,
                         int* __restrict__ nbuf) {
    __shared__ int red[256];
    const int b = blockIdx.x;
    const int t = threadIdx.x;
    int sum = 0;
    for (int s = t; s < S_; s += 256)
        sum += (mask[(size_t)b * S_ + s] != 0) ? 1 : 0;
    red[t] = sum;
    __syncthreads();
    for (int off = 128; off > 0; off >>= 1) {
        if (t < off) red[t] += red[t + off];
        __syncthreads();
    }
    if (t == 0) nbuf[b] = red[0];
}

// ---------------------------------------------------------------------------
// Pass 2: fused gather + GEMM + bias.
// out[b,s,o] = sum_k fused[b,s,k] * W[o,k] + bias[o]
// fused[b,s,k] = k < D ? forward[b,s,k]
//                      : reverse[b, (s<n_b ? n_b-1-s : s), k-D]
// 64x64 output tile per 256-thread (8-wave) workgroup, K-step 64,
// V_WMMA_F32_16X16X4_F32 accumulation from LDS.
// ---------------------------------------------------------------------------
__global__ __launch_bounds__(256)
void profam_gemm_kernel(const float* __restrict__ fwd,
                        const float* __restrict__ rev,
                        const float* __restrict__ W,
                        const float* __restrict__ bias,
                        const int*   __restrict__ nbuf,
                        float* __restrict__ out) {
    __shared__ float As[TM * LDP];
    __shared__ float Bs[TN * LDP];

    const int n0 = blockIdx.x * TN;            // output-column tile base
    const int m0 = blockIdx.y * TM;            // fused-row tile base (S % TM == 0)
    const int b  = m0 / S_;                    // whole tile lies in one batch
    const int nvalid = nbuf[b];

    const int t    = threadIdx.x;
    const int wave = t >> 5;
    const int lane = t & 31;

    // --- cooperative load coordinates: thread -> (row, 16-wide col group) ---
    const int lr = t >> 2;                     // 0..63
    const int lc = (t & 3) * 16;               // 0,16,32,48

    const int s_row = (m0 % S_) + lr;          // sequence position of A row
    const int s_rev = (s_row < nvalid) ? (nvalid - 1 - s_row) : s_row;
    const float* arow_f = fwd + ((size_t)b * S_ + s_row) * D_;
    const float* arow_r = rev + ((size_t)b * S_ + s_rev) * D_;
    const float* brow   = W   + (size_t)(n0 + lr) * K2_;   // W[o, :], k-contiguous

    // --- wave -> (M subtile, two N subtiles) ---
    const int mi  = wave & 3;                  // 0..3
    const int ni0 = (wave >> 2) * 2;           // 0 or 2
    const int ni1 = ni0 + 1;

    v8f acc0 = {};
    v8f acc1 = {};

    // Fragment read bases per ISA 16x4 f32 A layout:
    // lanes 0-15: (row = lane, K = kk+0/kk+1); lanes 16-31: (row = lane-16, K = kk+2/kk+3)
    const int halfSel = lane >> 4;             // 0 | 1
    const int rsub    = lane & 15;             // 0..15
    const float* aPtr  = As + (mi  * 16 + rsub) * LDP + 2 * halfSel;
    const float* b0Ptr = Bs + (ni0 * 16 + rsub) * LDP + 2 * halfSel;
    const float* b1Ptr = Bs + (ni1 * 16 + rsub) * LDP + 2 * halfSel;

    for (int k0 = 0; k0 < K2_; k0 += TK) {
        // ---- stage A tile: gathered fused activations, 4x float4 per thread ----
        const float* srcA = (k0 < D_) ? (arow_f + k0) : (arow_r + (k0 - D_));
        float* adst = As + lr * LDP + lc;
        #pragma unroll
        for (int j = 0; j < 16; j += 4)
            *(float4*)(adst + j) = *(const float4*)(srcA + lc + j);

        // ---- stage B tile: W rows (k-contiguous), 4x float4 per thread ----
        const float* srcB = brow + k0 + lc;
        float* bdst = Bs + lr * LDP + lc;
        #pragma unroll
        for (int j = 0; j < 16; j += 4)
            *(float4*)(bdst + j) = *(const float4*)(srcB + j);

        // ---- prefetch next K panel into L2 (global_prefetch_b8) ----
        const int kn = (k0 + TK < K2_) ? (k0 + TK) : k0;
        const float* pfA = (kn < D_) ? (arow_f + kn) : (arow_r + (kn - D_));
        __builtin_prefetch(pfA + lc, 0, 0);
        __builtin_prefetch(brow + kn + lc, 0, 0);

        __syncthreads();

        // ---- 16 K-slabs of 4: one ds_load_b64 per operand, 2 WMMAs sharing A ----
        #pragma unroll
        for (int kk = 0; kk < TK; kk += 4) {
            v2f a  = *(const v2f*)(aPtr  + kk);
            v2f bb0 = *(const v2f*)(b0Ptr + kk);
            v2f bb1 = *(const v2f*)(b1Ptr + kk);
            acc0 = __builtin_amdgcn_wmma_f32_16x16x4_f32(
                false, a, false, bb0, (short)0, acc0, false, false);
            acc1 = __builtin_amdgcn_wmma_f32_16x16x4_f32(
                false, a, false, bb1, (short)0, acc1, false, false);
        }
        __syncthreads();
    }

    // ---- epilogue: C/D layout -> VGPR j holds (M = j + 8*halfSel, N = rsub) ----
    const int ncol0 = n0 + ni0 * 16 + rsub;
    const int ncol1 = n0 + ni1 * 16 + rsub;
    const float bias0 = bias[ncol0];
    const float bias1 = bias[ncol1];
    const int mrowBase = m0 + mi * 16 + halfSel * 8;
    #pragma unroll
    for (int j = 0; j < 8; ++j) {
        const size_t row = (size_t)(mrowBase + j);
        out[row * OUT_ + ncol0] = acc0[j] + bias0;
        out[row * OUT_ + ncol1] = acc1[j] + bias1;
    }
}

// ---------------------------------------------------------------------------
// Harness entry point.
// d_in order: forward_repr[B,S,D] f32, reverse_repr[B,S,D] f32,
//             mask[B,S] bool, W[OUT,2D] f32, b[OUT] f32
// d_out: [B,S,OUT] f32
// ---------------------------------------------------------------------------
extern "C" void kernel_launch(void* const* d_in, const int* in_sizes, int n_in,
                              void* d_out, int out_size, void* d_ws, size_t ws_size,
                              hipStream_t stream) {
    const float*         fwd  = (const float*)d_in[0];
    const float*         rv   = (const float*)d_in[1];
    const unsigned char* mask = (const unsigned char*)d_in[2];
    const float*         W    = (const float*)d_in[3];
    const float*         bias = (const float*)d_in[4];
    float* out = (float*)d_out;
    int*   nbuf = (int*)d_ws;              // B_ ints of scratch

    profam_count_kernel<<<B_, 256, 0, stream>>>(mask, nbuf);

    dim3 grid(OUT_ / TN, (B_ * S_) / TM);  // (20, 256)
    profam_gemm_kernel<<<grid, 256, 0, stream>>>(fwd, rv, W, bias, nbuf, out);
}